// TrigoLinear_79791902425667
// MI455X (gfx1250) — compile-verified
//
#include <hip/hip_runtime.h>
#include <hip/hip_bf16.h>

// Problem constants (match reference).
constexpr int B_    = 1024;
constexpr int IN_   = 512;
constexpr int OUT_  = 512;
constexpr int TB    = 32;           // batch tile (halves weight/bias L2 re-reads vs 16)
constexpr int TO    = 16;           // output tile
constexpr int NTHR  = TB * TO;      // 512 threads = 16 waves (wave32)
constexpr int CHUNK = 128;          // i-chunk
constexpr int PITCH = CHUNK + 4;    // +4 dwords -> 4-bank rotation per row, 16B-aligned rows
constexpr float INV2PI = 0.15915494309189535f; // v_sin_f32 takes revolutions

// ---- CDNA5 async global->LDS path (guarded; fallback keeps compile clean) ----
#if defined(__HIP_DEVICE_COMPILE__) && defined(__has_builtin)
#if __has_builtin(__builtin_amdgcn_global_load_async_to_lds_b128)
#define HAS_ASYNC_LDS 1
#else
#define HAS_ASYNC_LDS 0
#endif
#if __has_builtin(__builtin_amdgcn_s_wait_asynccnt)
#define HAS_WAIT_ASYNC 1
#else
#define HAS_WAIT_ASYNC 0
#endif
#else
#define HAS_ASYNC_LDS 0
#define HAS_WAIT_ASYNC 0
#endif

// Exact builtin param types: v4i in AS(1)/AS(3) (clang prints AS1 as `__device__`).
typedef int v4i_t __attribute__((ext_vector_type(4)));
typedef __attribute__((address_space(1))) v4i_t glb_v4i;
typedef __attribute__((address_space(3))) v4i_t lds_v4i;

__device__ __forceinline__ void async_copy16(const float* g, float* l) {
#if HAS_ASYNC_LDS
    // global_load_async_to_lds_b128: (global v4i*, lds v4i*, imm offset, imm cpol)
    __builtin_amdgcn_global_load_async_to_lds_b128((glb_v4i*)g, (lds_v4i*)l, 0, 0);
#else
    *reinterpret_cast<float4*>(l) = *reinterpret_cast<const float4*>(g);
#endif
}

__device__ __forceinline__ void wait_async0() {
#if HAS_ASYNC_LDS
#if HAS_WAIT_ASYNC
    __builtin_amdgcn_s_wait_asynccnt(0);
#else
    asm volatile("s_wait_asynccnt 0" ::: "memory");
#endif
#endif
}

__global__ __launch_bounds__(NTHR)
void trigo_linear_kernel(const float* __restrict__ x,
                         const float* __restrict__ weight,
                         const float* __restrict__ bias,
                         float* __restrict__ out) {
    // x tile [32][PITCH] + 3 weight tiles [16][PITCH]: 42.2 KB
    __shared__ __align__(16) float sm[(TB + 3 * TO) * PITCH];
    float* sx  = sm;                         // [TB][PITCH]
    float* sws = sm + TB * PITCH;            // [TO][PITCH]  w_sin * inv2pi
    float* sbs = sm + (TB + TO) * PITCH;     // [TO][PITCH]  b_sin * inv2pi
    float* swo = sm + (TB + 2 * TO) * PITCH; // [TO][PITCH]  w_out

    const int tid = threadIdx.x;  // 512 threads
    // Lane mapping: within any wave, lanes 0-15 and 16-31 read the SAME 16 x-rows
    // (LDS broadcast) while covering two o values -> conflict-free b128 reads.
    const int bl = (tid & 15) | ((tid >> 8) << 4);   // 0..31
    const int ol = (tid >> 4) & 15;                  // 0..15
    const int b0 = blockIdx.x * TB;
    const int o0 = blockIdx.y * TO;

    float a0 = 0.f, a1 = 0.f, a2 = 0.f, a3 = 0.f;

    for (int c = 0; c < IN_ / CHUNK; ++c) {
        const int i0 = c * CHUNK;
        if (c) __syncthreads();   // everyone done reading previous tiles

        // ---- x tile: 32 rows x 128 f32 = 1024 async b128 transfers, 2/thread ----
        #pragma unroll
        for (int k = 0; k < (TB * CHUNK / 4) / NTHR; ++k) {
            int idx = tid + k * NTHR;           // 0..1023
            int row = idx >> 5;                 // / (CHUNK/4)
            int c4  = idx & 31;
            async_copy16(x + (size_t)(b0 + row) * IN_ + i0 + c4 * 4,
                         sx + row * PITCH + c4 * 4);
        }

        // ---- weight tile: interleaved (w_out, w_sin) pairs; b128 covers 2 i's ----
        // Prescale w_sin by 1/(2pi): inner loop is pure fma -> v_sin -> fma.
        #pragma unroll
        for (int k = 0; k < (TO * CHUNK / 2) / NTHR; ++k) {
            int idx = tid + k * NTHR;           // 0..1023
            int o   = idx >> 6;                 // / 64 pairs per row
            int p   = idx & 63;
            int i   = p * 2;
            const float4 wv = *reinterpret_cast<const float4*>(
                weight + ((size_t)(o0 + o) * IN_ + i0 + i) * 2);
            float2 wo2; wo2.x = wv.x;           wo2.y = wv.z;
            float2 ws2; ws2.x = wv.y * INV2PI;  ws2.y = wv.w * INV2PI;
            *reinterpret_cast<float2*>(swo + o * PITCH + i) = wo2;
            *reinterpret_cast<float2*>(sws + o * PITCH + i) = ws2;
        }

        // ---- b_sin tile: rows stride 513 (odd) -> b32 loads, prescaled ----
        #pragma unroll
        for (int k = 0; k < (TO * CHUNK) / NTHR; ++k) {
            int idx = tid + k * NTHR;           // 0..2047
            int o   = idx >> 7;                 // / CHUNK
            int i   = idx & (CHUNK - 1);
            sbs[o * PITCH + i] =
                bias[(size_t)(o0 + o) * (IN_ + 1) + i0 + i] * INV2PI;
        }

        wait_async0();       // ASYNCcnt==0: our async b128s landed in LDS
        __syncthreads();     // + dscnt fence for the staged stores

        // ---- compute: per 4 elems = 4x ds_load_b128, 8 FMA, 4 v_sin_f32 ----
        const float4* xr  = reinterpret_cast<const float4*>(sx  + bl * PITCH);
        const float4* wsr = reinterpret_cast<const float4*>(sws + ol * PITCH);
        const float4* bsr = reinterpret_cast<const float4*>(sbs + ol * PITCH);
        const float4* wor = reinterpret_cast<const float4*>(swo + ol * PITCH);

        #pragma unroll 8
        for (int j = 0; j < CHUNK / 4; ++j) {
            float4 xv = xr[j], wv = wsr[j], bv = bsr[j], ov = wor[j];
            // v_sin_f32 input is in revolutions; 1/(2pi) folded into wv/bv.
            a0 = fmaf(ov.x, __builtin_amdgcn_sinf(fmaf(xv.x, wv.x, bv.x)), a0);
            a1 = fmaf(ov.y, __builtin_amdgcn_sinf(fmaf(xv.y, wv.y, bv.y)), a1);
            a2 = fmaf(ov.z, __builtin_amdgcn_sinf(fmaf(xv.z, wv.z, bv.z)), a2);
            a3 = fmaf(ov.w, __builtin_amdgcn_sinf(fmaf(xv.w, wv.w, bv.w)), a3);
        }
    }

    const float bout = bias[(size_t)(o0 + ol) * (IN_ + 1) + IN_];
    out[(size_t)(b0 + bl) * OUT_ + (o0 + ol)] = (a0 + a1) + (a2 + a3) + bout;
}

extern "C" void kernel_launch(void* const* d_in, const int* in_sizes, int n_in,
                              void* d_out, int out_size, void* d_ws, size_t ws_size,
                              hipStream_t stream) {
    (void)in_sizes; (void)n_in; (void)out_size; (void)d_ws; (void)ws_size;
    const float* x      = (const float*)d_in[0];   // (1024, 512) f32
    const float* weight = (const float*)d_in[1];   // (512, 512, 2) f32
    const float* bias   = (const float*)d_in[2];   // (512, 513) f32
    float*       out    = (float*)d_out;           // (1024, 512) f32

    dim3 grid(B_ / TB, OUT_ / TO);   // 32 x 32 = 1024 blocks
    trigo_linear_kernel<<<grid, dim3(NTHR), 0, stream>>>(x, weight, bias, out);
}